// ExpEig_52553219834314
// MI455X (gfx1250) — compile-verified
//
#include <hip/hip_runtime.h>
#include <hip/hip_bf16.h>

typedef float v2f __attribute__((ext_vector_type(2)));
typedef float v8f __attribute__((ext_vector_type(8)));

#define LSTRIDE 34            // row stride in dwords: conflict-free + even (8B align)
#define MSIZE   (32 * LSTRIDE)
#define WAVES_PER_BLOCK 4

__device__ __forceinline__ void lds_fence() {
    // DS ops are in-order per wave; this stops the compiler from reordering
    // LDS accesses across the fence and drains outstanding DS ops.
    asm volatile("s_wait_dscnt 0" ::: "memory");
}

// C = A * B for 32x32 fp32 matrices in LDS (row-major, stride LSTRIDE dwords).
// One full wave32; uses V_WMMA_F32_16X16X4_F32, 2x2 C-tiles x 8 K-chunks.
__device__ __forceinline__ void mm32(const float* __restrict__ A,
                                     const float* __restrict__ B,
                                     float* __restrict__ C,
                                     int lane) {
    const int half = lane >> 4;       // 0: lanes 0-15, 1: lanes 16-31
    const int l16  = lane & 15;

    v8f c00 = {}, c01 = {}, c10 = {}, c11 = {};

#pragma unroll
    for (int kb = 0; kb < 8; ++kb) {
        const int k0 = kb * 4 + half * 2;   // K base for this lane-half

        // A fragments (16x4, MxK): a[j] = A[m][k0+j], m = tile*16 + l16
        v2f a0, a1, b0, b1;
        a0[0] = A[l16 * LSTRIDE + k0];
        a0[1] = A[l16 * LSTRIDE + k0 + 1];
        a1[0] = A[(16 + l16) * LSTRIDE + k0];
        a1[1] = A[(16 + l16) * LSTRIDE + k0 + 1];

        // B fragments (4x16, KxN): b[j] = B[k0+j][n], n = tile*16 + l16
        b0[0] = B[k0 * LSTRIDE + l16];
        b0[1] = B[(k0 + 1) * LSTRIDE + l16];
        b1[0] = B[k0 * LSTRIDE + 16 + l16];
        b1[1] = B[(k0 + 1) * LSTRIDE + 16 + l16];

        c00 = __builtin_amdgcn_wmma_f32_16x16x4_f32(false, a0, false, b0, (short)0, c00, false, false);
        c01 = __builtin_amdgcn_wmma_f32_16x16x4_f32(false, a0, false, b1, (short)0, c01, false, false);
        c10 = __builtin_amdgcn_wmma_f32_16x16x4_f32(false, a1, false, b0, (short)0, c10, false, false);
        c11 = __builtin_amdgcn_wmma_f32_16x16x4_f32(false, a1, false, b1, (short)0, c11, false, false);
    }

    // C/D layout: VGPR v -> row (half*8 + v) within the 16-row tile, col = l16
#pragma unroll
    for (int v = 0; v < 8; ++v) {
        const int r = half * 8 + v;
        C[r * LSTRIDE + l16]             = c00[v];
        C[r * LSTRIDE + 16 + l16]        = c01[v];
        C[(16 + r) * LSTRIDE + l16]      = c10[v];
        C[(16 + r) * LSTRIDE + 16 + l16] = c11[v];
    }
}

// expm(X) per 32x32 symmetric matrix via scaling-and-squaring + degree-10
// Taylor (Horner). One wave per matrix, WAVES_PER_BLOCK waves per block.
__global__ void __launch_bounds__(32 * WAVES_PER_BLOCK)
expm32_kernel(const float* __restrict__ x, float* __restrict__ out, int nmat) {
    __shared__ float smem[WAVES_PER_BLOCK * 3 * MSIZE];

    const int lane = threadIdx.x & 31;
    const int wid  = threadIdx.x >> 5;
    const int mat  = blockIdx.x * WAVES_PER_BLOCK + wid;
    if (mat >= nmat) return;    // uniform per wave

    float* X = smem + wid * 3 * MSIZE;
    float* P = X + MSIZE;
    float* T = P + MSIZE;

    const float* g = x + (size_t)mat * 1024;
    float*       o = out + (size_t)mat * 1024;

    // ---- Load (coalesced: lane = column) + column abs-sum (= inf-norm, symmetric) ----
    float csum = 0.0f;
#pragma unroll
    for (int r = 0; r < 32; ++r) {
        float v = g[r * 32 + lane];
        X[r * LSTRIDE + lane] = v;
        csum += fabsf(v);
    }
    // wave-wide max reduce -> uniform
    float nrm = csum;
#pragma unroll
    for (int off = 16; off > 0; off >>= 1)
        nrm = fmaxf(nrm, __shfl_xor(nrm, off, 32));

    const float theta = 0.5f;
    int s = 0;
    if (nrm > theta) {
        s = (int)ceilf(log2f(nrm / theta));
        if (s < 0)  s = 0;
        if (s > 24) s = 24;
    }
    const float scale = exp2f(-(float)s);

    // ---- Scale X in place, and init P = I + X/10 (Horner seed, degree m=10) ----
    lds_fence();
#pragma unroll
    for (int r = 0; r < 32; ++r) {
        float v = X[r * LSTRIDE + lane] * scale;
        X[r * LSTRIDE + lane] = v;
        P[r * LSTRIDE + lane] = ((r == lane) ? 1.0f : 0.0f) + v * 0.1f;
    }
    lds_fence();

    // ---- Horner: for k = 9..1: T = X*P ; P = I + T/k ----
    for (int k = 9; k >= 1; --k) {
        mm32(X, P, T, lane);
        lds_fence();
        const float inv = 1.0f / (float)k;
#pragma unroll
        for (int r = 0; r < 32; ++r)
            P[r * LSTRIDE + lane] =
                ((r == lane) ? 1.0f : 0.0f) + T[r * LSTRIDE + lane] * inv;
        lds_fence();
    }

    // ---- Squaring phase: P <- P^2, s times (ping-pong P/T) ----
    float* src = P;
    float* dst = T;
    for (int i = 0; i < s; ++i) {
        mm32(src, src, dst, lane);
        lds_fence();
        float* tmp = src; src = dst; dst = tmp;
    }

    // ---- Store (coalesced) ----
#pragma unroll
    for (int r = 0; r < 32; ++r)
        o[r * 32 + lane] = src[r * LSTRIDE + lane];
}

extern "C" void kernel_launch(void* const* d_in, const int* in_sizes, int n_in,
                              void* d_out, int out_size, void* d_ws, size_t ws_size,
                              hipStream_t stream) {
    (void)n_in; (void)d_ws; (void)ws_size; (void)out_size;
    const float* x = (const float*)d_in[0];
    float* out = (float*)d_out;
    const int nmat = in_sizes[0] / (32 * 32);   // 32768
    const int blocks = (nmat + WAVES_PER_BLOCK - 1) / WAVES_PER_BLOCK;
    expm32_kernel<<<blocks, 32 * WAVES_PER_BLOCK, 0, stream>>>(x, out, nmat);
}